// LennardJones_50697793962073
// MI455X (gfx1250) — compile-verified
//
#include <hip/hip_runtime.h>
#include <stdint.h>

typedef __attribute__((ext_vector_type(4))) float f32x4;
typedef __attribute__((ext_vector_type(4))) int   i32x4;

#define TPB 256

__global__ void lj_zero_kernel(float* __restrict__ out, int n) {
  int i = blockIdx.x * blockDim.x + threadIdx.x;
  if (i < n) out[i] = 0.0f;
}

__global__ __launch_bounds__(TPB) void lj_edges_kernel(
    const float* __restrict__ sigma, const float* __restrict__ delta,
    const float* __restrict__ epsilon,
    const float* __restrict__ elen, const float* __restrict__ ecut,
    const int* __restrict__ esrc, const int* __restrict__ edst,
    const int* __restrict__ types, float* __restrict__ out, int E)
{
  // Raw staged tables (sigma | delta | epsilon), then symmetrized+relu tables.
  __shared__ __align__(16) float stage[768];
  __shared__ float t_sig[256];
  __shared__ float t_dlt[256];
  __shared__ float t_eps[256];

  const int tid = threadIdx.x;

  // --- CDNA5 async global->LDS staging of the 3 x 16x16 f32 tables -------
  // 64 lanes x 16B per table = 1KB each. Tracked with ASYNCcnt.
  if (tid < 64) {
    uint32_t l_sig = (uint32_t)(uintptr_t)(&stage[tid * 4]); // low 32b of a
    uint32_t l_dlt = l_sig + 1024;                           // generic LDS ptr
    uint32_t l_eps = l_sig + 2048;                           // = LDS byte off
    uint64_t g_sig = (uint64_t)(uintptr_t)(sigma   + tid * 4);
    uint64_t g_dlt = (uint64_t)(uintptr_t)(delta   + tid * 4);
    uint64_t g_eps = (uint64_t)(uintptr_t)(epsilon + tid * 4);
    asm volatile("global_load_async_to_lds_b128 %0, %1, off"
                 :: "v"(l_sig), "v"(g_sig) : "memory");
    asm volatile("global_load_async_to_lds_b128 %0, %1, off"
                 :: "v"(l_dlt), "v"(g_dlt) : "memory");
    asm volatile("global_load_async_to_lds_b128 %0, %1, off"
                 :: "v"(l_eps), "v"(g_eps) : "memory");
  }
  asm volatile("s_wait_asynccnt 0x0" ::: "memory");
  __syncthreads();

  // --- Build symmetrized + relu'd pair tables: sym[i][j] = relu(P[min][max])
  {
    const int i = tid >> 4, j = tid & 15;
    const int r = (i < j) ? i : j;
    const int c = (i < j) ? j : i;
    const int f = r * 16 + c;
    t_sig[tid] = fmaxf(stage[f],       0.0f);
    t_dlt[tid] = fmaxf(stage[256 + f], 0.0f);
    t_eps[tid] = fmaxf(stage[512 + f], 0.0f);
  }
  __syncthreads();

  // --- Streamed edge loop: 4 edges per thread via NT b128 loads ----------
  const int nvec = E >> 2;
  for (int v = blockIdx.x * blockDim.x + tid; v < nvec;
       v += gridDim.x * blockDim.x) {
    i32x4 s4 = __builtin_nontemporal_load((const i32x4*)esrc + v);
    i32x4 d4 = __builtin_nontemporal_load((const i32x4*)edst + v);
    f32x4 l4 = __builtin_nontemporal_load((const f32x4*)elen + v);
    f32x4 c4 = __builtin_nontemporal_load((const f32x4*)ecut + v);
#pragma unroll
    for (int k = 0; k < 4; ++k) {
      const int   sa = s4[k];
      const int   p  = types[sa] * 16 + types[d4[k]];
      const float t  = t_sig[p] / (l4[k] - t_dlt[p]);
      const float t2 = t * t;
      const float x  = t2 * t2 * t2;                     // (sig/(r-d))^6
      const float eng = 2.0f * t_eps[p] * (x * x - x) * c4[k];
      atomicAdd(&out[sa], eng);                          // no-return f32 atomic
    }
  }

  // --- Scalar tail if E % 4 != 0 (not hit for E = 3.2M, kept for safety) --
  const int base = nvec << 2;
  if (blockIdx.x == 0) {
    for (int e = base + tid; e < E; e += TPB) {
      const int   sa = esrc[e];
      const int   p  = types[sa] * 16 + types[edst[e]];
      const float t  = t_sig[p] / (elen[e] - t_dlt[p]);
      const float t2 = t * t;
      const float x  = t2 * t2 * t2;
      atomicAdd(&out[sa], 2.0f * t_eps[p] * (x * x - x) * ecut[e]);
    }
  }
}

extern "C" void kernel_launch(void* const* d_in, const int* in_sizes, int n_in,
                              void* d_out, int out_size, void* d_ws, size_t ws_size,
                              hipStream_t stream) {
  // setup_inputs order: sigma, delta, epsilon, edge_len, edge_cutoff,
  //                     edge_index(2,E), atom_types(N)
  const float* sigma   = (const float*)d_in[0];
  const float* delta   = (const float*)d_in[1];
  const float* epsilon = (const float*)d_in[2];
  const float* elen    = (const float*)d_in[3];
  const float* ecut    = (const float*)d_in[4];
  const int*   eidx    = (const int*)d_in[5];
  const int*   types   = (const int*)d_in[6];
  float*       out     = (float*)d_out;

  const int E = in_sizes[3];   // edge count
  const int N = out_size;      // node count ([N,1] f32 output)

  lj_zero_kernel<<<(N + TPB - 1) / TPB, TPB, 0, stream>>>(out, N);

  int nvec   = E >> 2;
  int blocks = (nvec + TPB - 1) / TPB;
  if (blocks < 1) blocks = 1;
  lj_edges_kernel<<<blocks, TPB, 0, stream>>>(
      sigma, delta, epsilon, elen, ecut,
      eidx, eidx + E, types, out, E);
}